// InterpolationBlock2D_Lin_69896297775288
// MI455X (gfx1250) — compile-verified
//
#include <hip/hip_runtime.h>

// CDNA5 / gfx1250, wave32.
// out[d][p] = sum_{k<3} nodal[d][ wrap(conn[cell_id[p]][k]-1) ] * sf[p][k]
//
// Mapped onto V_WMMA_F32_16X16X4_F32: one 16-point tile per wave.
//   A (16x4, per component d): row m = gathered nodal values of point m.
//     Layout (ISA 7.12.2): lanes 0-15 hold {K0,K1}, lanes 16-31 hold {K2,K3}.
//   B (4x16): col n = shape functions of point n, same half-wave K split.
//   D[m][n] = sum_k G[m,k]*sf[n,k];  diagonal D[m][m] is the answer.
// Diagonal lives at (lane=m, vgpr=m) for m<8 and (lane=m+16, vgpr=m-8) for
// m>=8, so vgpr index == lane&7 on the valid lanes {0..7, 24..31}.

typedef __attribute__((ext_vector_type(2))) float v2f;
typedef __attribute__((ext_vector_type(8))) float v8f;

__device__ __forceinline__ float sel8(v8f c, unsigned k) {
    // 8-way select -> 7 v_cndmask, no scratch
    float t0 = (k & 1u) ? c[1] : c[0];
    float t1 = (k & 1u) ? c[3] : c[2];
    float t2 = (k & 1u) ? c[5] : c[4];
    float t3 = (k & 1u) ? c[7] : c[6];
    float u0 = (k & 2u) ? t1 : t0;
    float u1 = (k & 2u) ? t3 : t2;
    return (k & 4u) ? u1 : u0;
}

__global__ __launch_bounds__(256) void interp2d_wmma_kernel(
    const int*   __restrict__ cell_id,   // [n]
    const float* __restrict__ nodal,     // [2][n_nodes]
    const float* __restrict__ sf,        // [n][3]
    const int*   __restrict__ conn,      // [n_elems][3]
    float*       __restrict__ out,       // [2][n]
    int n, int n_nodes)
{
    const unsigned lane = threadIdx.x & 31u;
    const unsigned wave = threadIdx.x >> 5;      // 8 waves / block
    const unsigned m    = lane & 15u;            // point-in-tile (both halves)
    const unsigned half = lane >> 4;             // 0: K=0,1   1: K=2(,3=pad)

    const long tile = (long)blockIdx.x * 8 + wave;
    const long p    = tile * 16 + m;
    const long pc   = (p < n) ? p : (long)(n - 1);   // clamp: keep EXEC full

    // streaming prefetch (global_prefetch_b8); speculative, OOB is dropped
    __builtin_prefetch(cell_id + pc + 4096, 0, 1);
    __builtin_prefetch(sf + 3 * pc + 12288, 0, 1);

    const int cell = cell_id[pc];

    v2f a0, a1, b;
    if (half == 0) {
        int i0 = conn[3 * cell + 0] - 1; if (i0 < 0) i0 += n_nodes;
        int i1 = conn[3 * cell + 1] - 1; if (i1 < 0) i1 += n_nodes;
        a0[0] = nodal[i0];
        a0[1] = nodal[i1];
        a1[0] = nodal[n_nodes + i0];
        a1[1] = nodal[n_nodes + i1];
        b[0]  = sf[3 * pc + 0];
        b[1]  = sf[3 * pc + 1];
    } else {
        int i2 = conn[3 * cell + 2] - 1; if (i2 < 0) i2 += n_nodes;
        a0[0] = nodal[i2];
        a0[1] = 0.0f;                    // K=3 pad
        a1[0] = nodal[n_nodes + i2];
        a1[1] = 0.0f;
        b[0]  = sf[3 * pc + 2];
        b[1]  = 0.0f;                    // K=3 pad
    }
    // branches reconverge here -> EXEC all-1s for WMMA (ISA 7.12 requirement)

    v8f c0 = {};
    v8f c1 = {};
    // 8 args: (neg_a, A, neg_b, B, c_mod, C, reuse_a, reuse_b)
    c0 = __builtin_amdgcn_wmma_f32_16x16x4_f32(false, a0, false, b,
                                               (short)0, c0, false, false);
    c1 = __builtin_amdgcn_wmma_f32_16x16x4_f32(false, a1, false, b,
                                               (short)0, c1, false, false);

    // diagonal extraction
    const unsigned k  = lane & 7u;
    const float d0 = sel8(c0, k);
    const float d1 = sel8(c1, k);

    const bool validlane = (lane < 8u) || (lane >= 24u);
    const long pm = tile * 16 + ((lane < 8u) ? lane : (long)(lane - 16u));
    if (validlane && pm < n) {
        out[pm]           = d0;   // component 0
        out[(long)n + pm] = d1;   // component 1
    }
}

extern "C" void kernel_launch(void* const* d_in, const int* in_sizes, int n_in,
                              void* d_out, int out_size, void* d_ws, size_t ws_size,
                              hipStream_t stream) {
    // inputs (setup_inputs order): 0=x (unused), 1=cell_id (int),
    // 2=nodal_values (f32, [2][n_nodes]), 3=shape_functions (f32, [n][3]),
    // 4=connectivity (int, [n_elems][3])
    const int*   cell_id = (const int*)d_in[1];
    const float* nodal   = (const float*)d_in[2];
    const float* sf      = (const float*)d_in[3];
    const int*   conn    = (const int*)d_in[4];
    float*       out     = (float*)d_out;

    const int n       = in_sizes[1];
    const int n_nodes = in_sizes[2] / 2;

    const long tiles  = ((long)n + 15) / 16;       // 16 points per wave
    const int  blocks = (int)((tiles + 7) / 8);    // 8 waves per block

    interp2d_wmma_kernel<<<blocks, 256, 0, stream>>>(cell_id, nodal, sf, conn,
                                                     out, n, n_nodes);
}